// ResGATNet_25658134626482
// MI455X (gfx1250) — compile-verified
//
#include <hip/hip_runtime.h>
#include <hip/hip_bf16.h>

// ---------------------------------------------------------------------------
// ResGATNet for MI455X (gfx1250, wave32).
// Dense GEMMs use V_WMMA_F32_16X16X4_F32 (full fp32 on the matrix pipes).
// Edge softmax/scatter phase uses native f32 global atomics (the real
// bottleneck on this graph: ~0.5 GB irregular traffic vs ~5.6 GFLOP dense).
// ---------------------------------------------------------------------------

typedef __attribute__((ext_vector_type(2))) float v2f;
typedef __attribute__((ext_vector_type(8))) float v8f;

#define IN_C  128
#define HIDC  64
#define NHEAD 4
#define OUTC  16
#define GN_EPS 1e-5f

// ---------------- helpers ----------------
__device__ __forceinline__ unsigned ord_f32(float f) {
  unsigned u = __float_as_uint(f);
  return (u & 0x80000000u) ? ~u : (u | 0x80000000u);
}
__device__ __forceinline__ float unord_f32(unsigned u) {
  return (u & 0x80000000u) ? __uint_as_float(u ^ 0x80000000u)
                           : __uint_as_float(~u);
}

// ---------------- generic fills ----------------
__global__ void fill_f32_k(float* p, float v, int n) {
  int i = blockIdx.x * blockDim.x + threadIdx.x;
  if (i < n) p[i] = v;
}
__global__ void fill_u32_k(unsigned* p, unsigned v, int n) {
  int i = blockIdx.x * blockDim.x + threadIdx.x;
  if (i < n) p[i] = v;
}
__global__ void bias_bcast_k(float* p, const float* __restrict__ b, int n, int C) {
  int i = blockIdx.x * blockDim.x + threadIdx.x;
  if (i < n) p[i] = b[i % C];
}

// ---------------- WMMA fp32 GEMM: C[M x Nout] = A[M x K] * W[K x Nout] -------
// One wave -> one 16x16 output tile. M must be a multiple of 16 (caller
// handles tail rows with gemm_tail_k). K multiple of 4, Nout multiple of 16.
__global__ void wmma_gemm_f32(const float* __restrict__ A,
                              const float* __restrict__ W,
                              float* __restrict__ C,
                              int M, int K, int Nout) {
  int wave   = blockIdx.x * (blockDim.x >> 5) + (threadIdx.x >> 5);
  int tilesN = Nout >> 4;
  int tiles  = (M >> 4) * tilesN;
  if (wave >= tiles) return;                 // wave-uniform exit (EXEC all-1s)
  int tm = wave / tilesN, tn = wave % tilesN;
  int lane = threadIdx.x & 31;
  int half = lane >> 4;                      // 0: K pair {0,1}; 1: K pair {2,3}
  int l16  = lane & 15;
  int rowA = tm * 16 + l16;
  int colB = tn * 16 + l16;
  v8f acc = {};
  for (int k = 0; k < K; k += 4) {
    int ka = k + half * 2;
    v2f a, b;
    a.x = A[(size_t)rowA * K + ka];
    a.y = A[(size_t)rowA * K + ka + 1];
    b.x = W[(size_t)ka * Nout + colB];
    b.y = W[(size_t)(ka + 1) * Nout + colB];
    acc = __builtin_amdgcn_wmma_f32_16x16x4_f32(false, a, false, b,
                                                (short)0, acc, false, false);
  }
  int rowD = tm * 16 + half * 8;
#pragma unroll
  for (int r = 0; r < 8; ++r)
    C[(size_t)(rowD + r) * Nout + colB] = acc[r];
}

// scalar tail for M % 16 rows (not launched when N % 16 == 0)
__global__ void gemm_tail_k(const float* __restrict__ A, const float* __restrict__ W,
                            float* __restrict__ C, int m0, int M, int K, int Nout) {
  int t = blockIdx.x * blockDim.x + threadIdx.x;
  int cnt = (M - m0) * Nout;
  if (t >= cnt) return;
  int r = m0 + t / Nout, c = t % Nout;
  float s = 0.f;
  for (int k = 0; k < K; ++k) s += A[(size_t)r * K + k] * W[(size_t)k * Nout + c];
  C[(size_t)r * Nout + c] = s;
}

// ---------------- per-node attention logits --------------------------------
// asrc[n,h] = sum_c h[n,h,c]*a_src[h,c] ; adst likewise
__global__ void alpha_k(const float* __restrict__ h,
                        const float* __restrict__ a_src,
                        const float* __restrict__ a_dst,
                        float* __restrict__ asrc, float* __restrict__ adst,
                        int N, int H, int CH) {
  int t = blockIdx.x * blockDim.x + threadIdx.x;
  if (t >= N * H) return;
  int n = t / H, hh = t % H;
  const float* hp = h + (size_t)n * H * CH + hh * CH;
  const float* ws = a_src + hh * CH;
  const float* wd = a_dst + hh * CH;
  float s1 = 0.f, s2 = 0.f;
  for (int c = 0; c < CH; ++c) {
    float v = hp[c];
    s1 += v * ws[c];
    s2 += v * wd[c];
  }
  asrc[t] = s1;
  adst[t] = s2;
}

// ---------------- edge passes ----------------------------------------------
// pass 1: e = leaky_relu(asrc[src]+adst[dst], 0.2); segment max over dst
__global__ void edge_max_k(const float* __restrict__ asrc,
                           const float* __restrict__ adst,
                           const int* __restrict__ ei,
                           unsigned* __restrict__ segmax,
                           float* __restrict__ ebuf, int E, int N, int H) {
  int t = blockIdx.x * blockDim.x + threadIdx.x;
  int total = (E + N) * H;
  if (t >= total) return;
  int h = t % H, e = t / H;
  int s, d;
  if (e < E) { s = ei[e]; d = ei[E + e]; } else { s = d = e - E; }
  float v = asrc[s * H + h] + adst[d * H + h];
  v = v > 0.f ? v : 0.2f * v;
  ebuf[t] = v;
  atomicMax(segmax + d * H + h, ord_f32(v));
}

// pass 2: ex = exp(e - m[dst]); segment sum over dst
__global__ void edge_expsum_k(float* __restrict__ ebuf,
                              const unsigned* __restrict__ segmax,
                              float* __restrict__ den,
                              const int* __restrict__ ei, int E, int N, int H) {
  int t = blockIdx.x * blockDim.x + threadIdx.x;
  int total = (E + N) * H;
  if (t >= total) return;
  int h = t % H, e = t / H;
  int d = (e < E) ? ei[E + e] : (e - E);
  float m  = unord_f32(segmax[d * H + h]);
  float ex = expf(ebuf[t] - m);
  ebuf[t] = ex;
  atomicAdd(den + d * H + h, ex);
}

// pass 3: out[dst] += h[src] * (ex / (den[dst]+1e-16)). block = (CH, 4 edges)
__global__ void edge_scatter_k(const float* __restrict__ hsrc,
                               const float* __restrict__ ebuf,
                               const float* __restrict__ den,
                               const int* __restrict__ ei,
                               float* __restrict__ out,
                               int E, int N, int H, int CH) {
  int c = threadIdx.x;
  int e = blockIdx.x * blockDim.y + threadIdx.y;
  if (e >= E + N) return;
  int s, d;
  if (e < E) { s = ei[e]; d = ei[E + e]; } else { s = d = e - E; }
  for (int h = 0; h < H; ++h) {
    float alpha = ebuf[e * H + h] / (den[d * H + h] + 1e-16f);
    atomicAdd(out + (size_t)d * H * CH + h * CH + c,
              hsrc[(size_t)s * H * CH + h * CH + c] * alpha);
  }
}

// ---------------- GraphNorm -------------------------------------------------
// column-wise sum and sum-of-squares over nodes; blockDim.x == C
__global__ void gn_reduce_k(const float* __restrict__ x,
                            float* __restrict__ sum, float* __restrict__ sum2,
                            int N, int C, int rowsPerBlock) {
  int col = threadIdx.x;
  int r0 = blockIdx.x * rowsPerBlock;
  int r1 = r0 + rowsPerBlock;
  if (r1 > N) r1 = N;
  float s = 0.f, s2 = 0.f;
  for (int r = r0; r < r1; ++r) {
    float v = x[(size_t)r * C + col];
    s += v;
    s2 += v * v;
  }
  atomicAdd(sum + col, s);
  atomicAdd(sum2 + col, s2);
}

// y = elu( w*(x - ms*mean)/sqrt(var+eps) + b )
__global__ void gn_norm_elu_k(const float* __restrict__ x,
                              const float* __restrict__ sum,
                              const float* __restrict__ sum2,
                              const float* __restrict__ w,
                              const float* __restrict__ b,
                              const float* __restrict__ ms,
                              float* __restrict__ y, int N, int C) {
  int i = blockIdx.x * blockDim.x + threadIdx.x;
  if (i >= N * C) return;
  int col = i % C;
  float invN  = 1.0f / (float)N;
  float mean  = sum[col] * invN;
  float m2    = sum2[col] * invN;
  float sc    = ms[col];
  float var   = m2 - 2.f * sc * mean * mean + sc * sc * mean * mean;
  float ctr   = x[i] - sc * mean;
  float v     = w[col] * ctr * rsqrtf(var + GN_EPS) + b[col];
  y[i] = v > 0.f ? v : expm1f(v);
}

// ---------------- head: [x | x1 | x2] @ head_W + head_b (WMMA, Nout=16) -----
__global__ void head_wmma_k(const float* __restrict__ x,
                            const float* __restrict__ x1,
                            const float* __restrict__ x2,
                            const float* __restrict__ hw,
                            const float* __restrict__ hb,
                            float* __restrict__ out, int M) {
  int wave = blockIdx.x * (blockDim.x >> 5) + (threadIdx.x >> 5);
  int tiles = M >> 4;
  if (wave >= tiles) return;
  int lane = threadIdx.x & 31;
  int half = lane >> 4;
  int l16  = lane & 15;
  int rowA = wave * 16 + l16;
  v8f acc = {};
  const float* As[3] = {x, x1, x2};
  const int    Ks[3] = {IN_C, NHEAD * HIDC, HIDC};
  int roff = 0;
  for (int seg = 0; seg < 3; ++seg) {
    const float* A = As[seg];
    int K = Ks[seg];
    for (int k = 0; k < K; k += 4) {
      int ka = k + half * 2;
      v2f a, b;
      a.x = A[(size_t)rowA * K + ka];
      a.y = A[(size_t)rowA * K + ka + 1];
      b.x = hw[(size_t)(roff + ka) * OUTC + l16];
      b.y = hw[(size_t)(roff + ka + 1) * OUTC + l16];
      acc = __builtin_amdgcn_wmma_f32_16x16x4_f32(false, a, false, b,
                                                  (short)0, acc, false, false);
    }
    roff += K;
  }
  int rowD = wave * 16 + half * 8;
  float bias = hb[l16];
#pragma unroll
  for (int r = 0; r < 8; ++r)
    out[(size_t)(rowD + r) * OUTC + l16] = acc[r] + bias;
}

__global__ void head_tail_k(const float* __restrict__ x,
                            const float* __restrict__ x1,
                            const float* __restrict__ x2,
                            const float* __restrict__ hw,
                            const float* __restrict__ hb,
                            float* __restrict__ out, int m0, int M) {
  int t = blockIdx.x * blockDim.x + threadIdx.x;
  int cnt = (M - m0) * OUTC;
  if (t >= cnt) return;
  int r = m0 + t / OUTC, c = t % OUTC;
  float s = hb[c];
  for (int k = 0; k < IN_C; ++k)        s += x [(size_t)r*IN_C + k]        * hw[(size_t)k*OUTC + c];
  for (int k = 0; k < NHEAD*HIDC; ++k)  s += x1[(size_t)r*NHEAD*HIDC + k]  * hw[(size_t)(IN_C+k)*OUTC + c];
  for (int k = 0; k < HIDC; ++k)        s += x2[(size_t)r*HIDC + k]        * hw[(size_t)(IN_C+NHEAD*HIDC+k)*OUTC + c];
  out[(size_t)r * OUTC + c] = s;
}

// ---------------------------------------------------------------------------
static inline int cdiv(int a, int b) { return (a + b - 1) / b; }

extern "C" void kernel_launch(void* const* d_in, const int* in_sizes, int n_in,
                              void* d_out, int out_size, void* d_ws, size_t ws_size,
                              hipStream_t stream) {
  const float* x       = (const float*)d_in[0];
  const int*   ei      = (const int*)  d_in[1];
  const float* W1      = (const float*)d_in[2];
  const float* a_src1  = (const float*)d_in[3];
  const float* a_dst1  = (const float*)d_in[4];
  const float* b1      = (const float*)d_in[5];
  const float* gn1_w   = (const float*)d_in[6];
  const float* gn1_b   = (const float*)d_in[7];
  const float* gn1_ms  = (const float*)d_in[8];
  const float* W2      = (const float*)d_in[9];
  const float* a_src2  = (const float*)d_in[10];
  const float* a_dst2  = (const float*)d_in[11];
  const float* b2      = (const float*)d_in[12];
  const float* gn2_w   = (const float*)d_in[13];
  const float* gn2_b   = (const float*)d_in[14];
  const float* gn2_ms  = (const float*)d_in[15];
  const float* head_W  = (const float*)d_in[16];
  const float* head_b  = (const float*)d_in[17];
  float*       out     = (float*)d_out;

  const int N = in_sizes[0] / IN_C;       // 50000
  const int E = in_sizes[1] / 2;          // 800000
  const int C1 = NHEAD * HIDC;            // 256
  const int C2 = HIDC;                    // 64
  const int ET = E + N;                   // edges + self loops

  // ---- workspace layout (floats) ----
  float* ws     = (float*)d_ws;
  float* h1     = ws;                               // N*C1 ; later aliased as x1
  float* gat    = ws + (size_t)N * C1;              // N*C1 region
  float* x1     = h1;                               // overwrites h1 after layer1
  float* h2     = gat;                              // overwrites gat1
  float* gat2   = gat + (size_t)N * C2;
  float* x2     = gat + (size_t)2 * N * C2;
  float* asrc   = ws + (size_t)2 * N * C1;          // N*NHEAD
  float* adst   = asrc + (size_t)N * NHEAD;
  unsigned* segmax = (unsigned*)(adst + (size_t)N * NHEAD);
  float* den    = (float*)segmax + (size_t)N * NHEAD;
  float* ebuf   = den + (size_t)N * NHEAD;          // ET*NHEAD
  float* gnsum  = ebuf + (size_t)ET * NHEAD;        // 256
  float* gnsum2 = gnsum + C1;                       // 256

  // ordered-uint encoding of -FLT_MAX: bits(-FLT_MAX)=0xFF7FFFFF (sign set)
  // -> ~0xFF7FFFFF = 0x00800000  (host-side constant; ord_f32 is device-only)
  const unsigned NEG_BIG = 0x00800000u;
  const int T = 256;
  const int Mmain = (N / 16) * 16;
  const int tail  = N - Mmain;

  // ================= layer 1: GATConv(128 -> 4x64) =================
  {
    int tiles = (Mmain / 16) * (C1 / 16);
    wmma_gemm_f32<<<cdiv(tiles, 4), 128, 0, stream>>>(x, W1, h1, Mmain, IN_C, C1);
    if (tail > 0)
      gemm_tail_k<<<cdiv(tail * C1, T), T, 0, stream>>>(x, W1, h1, Mmain, N, IN_C, C1);

    alpha_k<<<cdiv(N * NHEAD, T), T, 0, stream>>>(h1, a_src1, a_dst1, asrc, adst, N, NHEAD, HIDC);

    bias_bcast_k<<<cdiv(N * C1, T), T, 0, stream>>>(gat, b1, N * C1, C1);
    fill_u32_k<<<cdiv(N * NHEAD, T), T, 0, stream>>>(segmax, NEG_BIG, N * NHEAD);
    fill_f32_k<<<cdiv(N * NHEAD, T), T, 0, stream>>>(den, 0.f, N * NHEAD);

    edge_max_k   <<<cdiv(ET * NHEAD, T), T, 0, stream>>>(asrc, adst, ei, segmax, ebuf, E, N, NHEAD);
    edge_expsum_k<<<cdiv(ET * NHEAD, T), T, 0, stream>>>(ebuf, segmax, den, ei, E, N, NHEAD);
    edge_scatter_k<<<cdiv(ET, 4), dim3(HIDC, 4), 0, stream>>>(h1, ebuf, den, ei, gat, E, N, NHEAD, HIDC);

    fill_f32_k<<<1, C1, 0, stream>>>(gnsum, 0.f, C1);
    fill_f32_k<<<1, C1, 0, stream>>>(gnsum2, 0.f, C1);
    gn_reduce_k<<<cdiv(N, 256), C1, 0, stream>>>(gat, gnsum, gnsum2, N, C1, 256);
    gn_norm_elu_k<<<cdiv(N * C1, T), T, 0, stream>>>(gat, gnsum, gnsum2,
                                                     gn1_w, gn1_b, gn1_ms, x1, N, C1);
  }

  // ================= layer 2: GATConv(256 -> 1x64) =================
  {
    int tiles = (Mmain / 16) * (C2 / 16);
    wmma_gemm_f32<<<cdiv(tiles, 4), 128, 0, stream>>>(x1, W2, h2, Mmain, C1, C2);
    if (tail > 0)
      gemm_tail_k<<<cdiv(tail * C2, T), T, 0, stream>>>(x1, W2, h2, Mmain, N, C1, C2);

    alpha_k<<<cdiv(N, T), T, 0, stream>>>(h2, a_src2, a_dst2, asrc, adst, N, 1, HIDC);

    bias_bcast_k<<<cdiv(N * C2, T), T, 0, stream>>>(gat2, b2, N * C2, C2);
    fill_u32_k<<<cdiv(N, T), T, 0, stream>>>(segmax, NEG_BIG, N);
    fill_f32_k<<<cdiv(N, T), T, 0, stream>>>(den, 0.f, N);

    edge_max_k   <<<cdiv(ET, T), T, 0, stream>>>(asrc, adst, ei, segmax, ebuf, E, N, 1);
    edge_expsum_k<<<cdiv(ET, T), T, 0, stream>>>(ebuf, segmax, den, ei, E, N, 1);
    edge_scatter_k<<<cdiv(ET, 4), dim3(HIDC, 4), 0, stream>>>(h2, ebuf, den, ei, gat2, E, N, 1, HIDC);

    fill_f32_k<<<1, C2, 0, stream>>>(gnsum, 0.f, C2);
    fill_f32_k<<<1, C2, 0, stream>>>(gnsum2, 0.f, C2);
    gn_reduce_k<<<cdiv(N, 256), C2, 0, stream>>>(gat2, gnsum, gnsum2, N, C2, 256);
    gn_norm_elu_k<<<cdiv(N * C2, T), T, 0, stream>>>(gat2, gnsum, gnsum2,
                                                     gn2_w, gn2_b, gn2_ms, x2, N, C2);
  }

  // ================= head: [x | x1 | x2] @ head_W + head_b =================
  {
    int tiles = Mmain / 16;
    head_wmma_k<<<cdiv(tiles, 4), 128, 0, stream>>>(x, x1, x2, head_W, head_b, out, Mmain);
    if (tail > 0)
      head_tail_k<<<cdiv(tail * OUTC, T), T, 0, stream>>>(x, x1, x2, head_W, head_b, out, Mmain, N);
  }
}